// CUD_Loss_25031069401432
// MI455X (gfx1250) — compile-verified
//
#include <hip/hip_runtime.h>
#include <hip/hip_bf16.h>

typedef float v2f __attribute__((ext_vector_type(2)));
typedef float v8f __attribute__((ext_vector_type(8)));

#define BATCH 8
#define IMH 512
#define IMW 512
#define HW (IMH * IMW)
#define NBINS 256
#define NLEV 5
#define PYR 349184  // 512^2 + 256^2 + 128^2 + 64^2 + 32^2 per image

__device__ __constant__ int LVL_OFF_c[NLEV] = {0, 262144, 327680, 344064, 348160};

// 1D Gaussian, sigma = 1.5, normalized (matches reference float64 window)
#define G0 0.1200784f
#define G1 0.2338807f
#define G2 0.2920813f

// Branch-free band lookup: gk[d] for d in [0,4], else 0 -> pure v_cndmask chain w/ literals
__device__ __forceinline__ float band_w(int d) {
    float r = 0.0f;
    r = (d == 0) ? G0 : r;
    r = (d == 1) ? G1 : r;
    r = (d == 2) ? G2 : r;
    r = (d == 3) ? G1 : r;
    r = (d == 4) ? G0 : r;
    return r;
}

// ---------------- RGB -> normalized LAB (CURL ImageProcessing.rgb_to_lab) ----------------
__device__ __forceinline__ float srgb_lin(float x) {
    return (x <= 0.04045f) ? x * (1.0f / 12.92f)
                           : powf((fmaxf(x, 1e-4f) + 0.055f) / 1.055f, 2.4f);
}
__device__ __forceinline__ float lab_f(float t) {
    const float eps = 6.0f / 29.0f;
    const float e3  = eps * eps * eps;
    return (t <= e3) ? (t / (3.0f * eps * eps) + 4.0f / 29.0f) : cbrtf(fmaxf(t, 1e-4f));
}
__device__ __forceinline__ void rgb2lab_norm(float r, float g, float b, float o[3]) {
    float lr = srgb_lin(r), lg = srgb_lin(g), lb = srgb_lin(b);
    float X = (lr * 0.412453f + lg * 0.357580f + lb * 0.180423f) * (1.0f / 0.950456f);
    float Y =  lr * 0.212671f + lg * 0.715160f + lb * 0.072169f;
    float Z = (lr * 0.019334f + lg * 0.119193f + lb * 0.950227f) * (1.0f / 1.088754f);
    float fx = lab_f(X), fy = lab_f(Y), fz = lab_f(Z);
    float L = 116.0f * fy - 16.0f;
    float A = 500.0f * (fx - fy);
    float B = 200.0f * (fy - fz);
    o[0] = __saturatef(L * 0.01f);
    o[1] = __saturatef((A / 110.0f + 1.0f) * 0.5f);
    o[2] = __saturatef((B / 110.0f + 1.0f) * 0.5f);
}
__device__ __forceinline__ int hist_bin(float v) {
    int i = (int)floorf(v * 256.0f);
    return (i < 0) ? 0 : (i > 255 ? 255 : i);
}
// low 32 bits of a generic pointer to LDS == LDS byte address
__device__ __forceinline__ unsigned lds_addr32(const void* p) {
    return (unsigned)(uintptr_t)p;
}

// ---------------- zero scratch accumulators ----------------
__global__ void k_zero(float* __restrict__ p, int n) {
    int i = blockIdx.x * blockDim.x + threadIdx.x;
    if (i < n) p[i] = 0.0f;
}

// ---------------- fused LAB + histogram + L1 + L-channel extract (single pass over 50 MB) ----------------
__global__ __launch_bounds__(256) void k_lab_hist(
    const float* __restrict__ pred, const float* __restrict__ tgt,
    float* __restrict__ predL, float* __restrict__ tgtL,
    float* __restrict__ histP, float* __restrict__ histT, float* __restrict__ l1sum)
{
    __shared__ unsigned hp[3 * NBINS], ht[3 * NBINS];
    __shared__ float red[256];
    const int tid = threadIdx.x;
    const int b   = blockIdx.y;
    for (int i = tid; i < 3 * NBINS; i += 256) { hp[i] = 0u; ht[i] = 0u; }
    __syncthreads();

    const float* pb = pred + (size_t)b * 3 * HW;
    const float* tb = tgt  + (size_t)b * 3 * HW;
    float lsum = 0.0f;
    #pragma unroll
    for (int i = 0; i < 4; ++i) {
        int p = blockIdx.x * 1024 + i * 256 + tid;
        __builtin_prefetch(pb + p + 2048, 0, 0);   // global_prefetch_b8 (streaming read-ahead)
        __builtin_prefetch(tb + p + 2048, 0, 0);
        float pl[3], tl[3];
        rgb2lab_norm(pb[p], pb[p + HW], pb[p + 2 * HW], pl);
        rgb2lab_norm(tb[p], tb[p + HW], tb[p + 2 * HW], tl);
        predL[(size_t)b * PYR + p] = pl[0];
        tgtL [(size_t)b * PYR + p] = tl[0];
        #pragma unroll
        for (int c = 0; c < 3; ++c) {
            lsum += fabsf(pl[c] - tl[c]);
            atomicAdd(&hp[c * NBINS + hist_bin(pl[c])], 1u);
            atomicAdd(&ht[c * NBINS + hist_bin(tl[c])], 1u);
        }
    }
    __syncthreads();
    for (int i = tid; i < 3 * NBINS; i += 256) {
        if (hp[i]) atomicAdd(&histP[b * 3 * NBINS + i], (float)hp[i]);
        if (ht[i]) atomicAdd(&histT[b * 3 * NBINS + i], (float)ht[i]);
    }
    red[tid] = lsum;
    __syncthreads();
    for (int s = 128; s > 0; s >>= 1) {
        if (tid < s) red[tid] += red[tid + s];
        __syncthreads();
    }
    if (tid == 0) atomicAdd(l1sum, red[0]);
}

// ---------------- SSIM per level ----------------------------------------------------------
// One wave (32 lanes) per 16x16 output tile.
//  * Halo staging: GLOBAL_LOAD_ASYNC_TO_LDS_B128 (ASYNCcnt) for in-bounds 16B groups,
//    predicated scalar fallback (zero SAME-padding) for boundary groups (disjoint LDS slots).
//  * Horizontal Gaussian tap in VALU on {x, y, x^2, y^2, xy} -> 20x16 each, in LDS.
//  * Vertical tap on the matrix pipe: out(16x16) = A_band(16x20) x hconv(20x16), decomposed
//    into 5 chained v_wmma_f32_16x16x4_f32 accumulations per signal (25 WMMAs / tile).
__global__ __launch_bounds__(32) void k_ssim(
    const float* __restrict__ predPyr, const float* __restrict__ tgtPyr,
    float* __restrict__ ssim_sum, float* __restrict__ cs_sum, int level)
{
    const int lane = threadIdx.x;
    const int b    = blockIdx.y;
    const int dim  = IMH >> level;
    const int tpr  = dim >> 4;
    const int tx   = blockIdx.x % tpr;
    const int ty   = blockIdx.x / tpr;
    const float* i1 = predPyr + (size_t)b * PYR + LVL_OFF_c[level];
    const float* i2 = tgtPyr  + (size_t)b * PYR + LVL_OFF_c[level];

    __shared__ float t1[20][20], t2[20][20];   // row stride 80 B -> every 4-float group 16B-aligned
    __shared__ float hb[5][20][17];            // stride 17: conflict-free across lane halves

    // ---- halo load: 20 rows x 5 groups of 4 floats, per image ----
    for (int idx = lane; idx < 100; idx += 32) {
        int rr = idx / 5, c4 = idx % 5;
        int y  = ty * 16 + rr - 2;
        int x0 = tx * 16 + c4 * 4 - 2;
        bool rowok = (y >= 0) && (y < dim);
        bool full  = rowok && (x0 >= 0) && (x0 + 3 < dim);
        if (full) {
            unsigned l1a  = lds_addr32(&t1[rr][c4 * 4]);
            unsigned l2a  = lds_addr32(&t2[rr][c4 * 4]);
            unsigned goff = (unsigned)((y * dim + x0) * 4);
            asm volatile("global_load_async_to_lds_b128 %0, %1, %2"
                         :: "v"(l1a), "v"(goff), "s"(i1) : "memory");
            asm volatile("global_load_async_to_lds_b128 %0, %1, %2"
                         :: "v"(l2a), "v"(goff), "s"(i2) : "memory");
        } else {
            #pragma unroll
            for (int u = 0; u < 4; ++u) {
                int x = x0 + u;
                bool ok = rowok && (x >= 0) && (x < dim);
                t1[rr][c4 * 4 + u] = ok ? i1[y * dim + x] : 0.0f;
                t2[rr][c4 * 4 + u] = ok ? i2[y * dim + x] : 0.0f;
            }
        }
    }
    asm volatile("s_wait_asynccnt 0x0" ::: "memory");
    __syncthreads();

    // ---- horizontal tap on the 5 signals -> 20x16 each (symmetric-weight factoring) ----
    for (int idx = lane; idx < 320; idx += 32) {
        int j = idx / 16, n = idx % 16;
        float a0 = t1[j][n],     a1 = t1[j][n + 1], a2 = t1[j][n + 2];
        float a3 = t1[j][n + 3], a4 = t1[j][n + 4];
        float c0 = t2[j][n],     c1 = t2[j][n + 1], c2 = t2[j][n + 2];
        float c3 = t2[j][n + 3], c4v = t2[j][n + 4];
        hb[0][j][n] = G0 * (a0 + a4) + G1 * (a1 + a3) + G2 * a2;
        hb[1][j][n] = G0 * (c0 + c4v) + G1 * (c1 + c3) + G2 * c2;
        hb[2][j][n] = G0 * (a0 * a0 + a4 * a4) + G1 * (a1 * a1 + a3 * a3) + G2 * a2 * a2;
        hb[3][j][n] = G0 * (c0 * c0 + c4v * c4v) + G1 * (c1 * c1 + c3 * c3) + G2 * c2 * c2;
        hb[4][j][n] = G0 * (a0 * c0 + a4 * c4v) + G1 * (a1 * c1 + a3 * c3) + G2 * a2 * c2;
    }
    __syncthreads();

    // ---- vertical tap via banded WMMA ----
    const int m    = lane & 15;   // A row / B,C,D column (ISA §7.12.2 f32 layouts)
    const int half = lane >> 4;   // K-phase {0,1} vs {2,3} within each 16x4 / 4x16 chunk
    v8f acc[5];
    #pragma unroll
    for (int s = 0; s < 5; ++s) { v8f z = {0, 0, 0, 0, 0, 0, 0, 0}; acc[s] = z; }

    #pragma unroll
    for (int k = 0; k < 5; ++k) {
        int k0 = 4 * k + (half ? 2 : 0);
        v2f av;                                   // A_band[m][j] = gk[j - m], j in [m, m+4]
        av.x = band_w(k0 - m);
        av.y = band_w(k0 + 1 - m);
        #pragma unroll
        for (int s = 0; s < 5; ++s) {
            v2f bv;
            bv.x = hb[s][k0][m];
            bv.y = hb[s][k0 + 1][m];
            acc[s] = __builtin_amdgcn_wmma_f32_16x16x4_f32(
                false, av, false, bv, (short)0, acc[s], false, false);
        }
    }

    // ---- per-lane: 8 elements of each statistic (C/D layout: M = v + 8*half, N = lane&15) ----
    const float C1 = 1e-4f, C2 = 9e-4f;
    float ssum = 0.0f, csum = 0.0f;
    #pragma unroll
    for (int v = 0; v < 8; ++v) {
        float mu1 = acc[0][v], mu2 = acc[1][v];
        float s1  = acc[2][v] - mu1 * mu1;
        float s2  = acc[3][v] - mu2 * mu2;
        float s12 = acc[4][v] - mu1 * mu2;
        float t2s  = 2.0f * s12 + C2;
        float den2 = s1 + s2 + C2;
        ssum += ((2.0f * mu1 * mu2 + C1) * t2s) / ((mu1 * mu1 + mu2 * mu2 + C1) * den2);
        csum += t2s / den2;
    }
    #pragma unroll
    for (int off = 16; off > 0; off >>= 1) {
        ssum += __shfl_xor(ssum, off, 32);
        csum += __shfl_xor(csum, off, 32);
    }
    if (lane == 0) {
        atomicAdd(&ssim_sum[b * NLEV + level], ssum);
        atomicAdd(&cs_sum  [b * NLEV + level], csum);
    }
}

// ---------------- 2x2 average pool (level -> level+1), both pyramids via grid.z ----------------
__global__ void k_pool(float* __restrict__ predPyr, float* __restrict__ tgtPyr, int level) {
    float* base = blockIdx.z ? tgtPyr : predPyr;
    const int od  = IMH >> (level + 1);
    const int id2 = od * 2;
    const int n   = od * od;
    int i = blockIdx.x * blockDim.x + threadIdx.x;
    if (i >= n) return;
    int y = i / od, x = i % od;
    int b = blockIdx.y;
    const float* in  = base + (size_t)b * PYR + LVL_OFF_c[level];
    float*       out = base + (size_t)b * PYR + LVL_OFF_c[level + 1];
    const float* r0 = in + (2 * y) * id2 + 2 * x;
    const float* r1 = r0 + id2;
    out[i] = 0.25f * (r0[0] + r0[1] + r1[0] + r1[1]);
}

// ---------------- final scalar combine ----------------
__global__ __launch_bounds__(256) void k_final(
    const float* __restrict__ histP, const float* __restrict__ histT,
    const float* __restrict__ accum, float* __restrict__ out)
{
    __shared__ float l1img[BATCH];
    int tid = threadIdx.x, g = tid >> 5, lane = tid & 31;
    float s = 0.0f;
    for (int i = lane; i < 3 * NBINS; i += 32)
        s += fabsf(histP[g * 3 * NBINS + i] - histT[g * 3 * NBINS + i]);
    #pragma unroll
    for (int off = 16; off > 0; off >>= 1) s += __shfl_xor(s, off, 32);
    if (lane == 0) l1img[g] = s / (float)(3 * NBINS);
    __syncthreads();
    if (tid == 0) {
        const float W[5] = {0.0448f, 0.2856f, 0.3001f, 0.2363f, 0.1333f};
        float hist_loss = 0.0f;
        for (int i = 0; i < BATCH; ++i)
            hist_loss += exp2f((float)(i - BATCH)) * (l1img[i] + 1.0f);
        const float scaler = (float)HW / 20.0f;
        hist_loss = hist_loss / (float)BATCH / scaler;
        float lab_l1 = accum[0] / (float)((size_t)BATCH * 3 * HW);
        out[0] = lab_l1 + hist_loss;

        float sl = 0.0f;
        for (int i = 0; i < BATCH; ++i) {
            float ms = 1.0f;
            for (int l = 0; l < NLEV - 1; ++l) {
                float npix = (float)((IMH >> l) * (IMW >> l));
                float cs = accum[41 + i * NLEV + l] / npix;
                ms *= powf((cs + 1.0f) * 0.5f, W[l]);
            }
            float np4 = (float)((IMH >> 4) * (IMW >> 4));
            float sm = accum[1 + i * NLEV + 4] / np4;
            ms *= powf((sm + 1.0f) * 0.5f, W[4]);
            sl += 1.0f - ms;
        }
        out[1] = sl / (float)BATCH;
    }
}

extern "C" void kernel_launch(void* const* d_in, const int* in_sizes, int n_in,
                              void* d_out, int out_size, void* d_ws, size_t ws_size,
                              hipStream_t stream) {
    const float* pred = (const float*)d_in[1];   // predicted_img_batch
    const float* tgt  = (const float*)d_in[2];   // target_img_batch
    float* out = (float*)d_out;                  // [lab_loss, ssim_loss]
    float* ws  = (float*)d_ws;

    // workspace layout (floats)
    float* predL = ws;                                    // BATCH * PYR
    float* tgtL  = predL + (size_t)BATCH * PYR;           // BATCH * PYR
    float* histP = tgtL  + (size_t)BATCH * PYR;           // BATCH * 768
    float* histT = histP + BATCH * 3 * NBINS;             // BATCH * 768
    float* accum = histT + BATCH * 3 * NBINS;             // [0]=l1sum, [1..40]=ssim, [41..80]=cs

    const int zn = 2 * BATCH * 3 * NBINS + 81;            // hists + accumulators (contiguous)
    k_zero<<<(zn + 255) / 256, 256, 0, stream>>>(histP, zn);

    k_lab_hist<<<dim3(HW / 1024, BATCH), 256, 0, stream>>>(
        pred, tgt, predL, tgtL, histP, histT, accum);

    for (int l = 0; l < NLEV; ++l) {
        int dim   = IMH >> l;
        int tiles = (dim / 16) * (dim / 16);
        k_ssim<<<dim3(tiles, BATCH), 32, 0, stream>>>(predL, tgtL, accum + 1, accum + 41, l);
        if (l < NLEV - 1) {
            int od = dim / 2;
            int n  = od * od;
            k_pool<<<dim3((n + 255) / 256, BATCH, 2), 256, 0, stream>>>(predL, tgtL, l);
        }
    }
    k_final<<<1, 256, 0, stream>>>(histP, histT, accum, out);
}